// ChamferL2Loss_87222195847748
// MI455X (gfx1250) — compile-verified
//
#include <hip/hip_runtime.h>
#include <cfloat>

// ---------------- problem constants (from reference) ----------------
#define NPTS    7000
#define NB      4
#define TILES_I 440                    // padded iteration count (divisible by 4)
#define TPAD    (TILES_I * 16)         // 7040 padded target columns
#define TILES_M ((NPTS + 15) / 16)     // 438 pred row tiles
#define BIGC    1.0e10f
#define NSORT   8192                   // next pow2 >= NPTS for bitonic sort

typedef float v2f __attribute__((ext_vector_type(2)));
typedef float v8f __attribute__((ext_vector_type(8)));

// ---------------- workspace layout (float offsets) ----------------
#define OFF_RLO   0                        // [NB][3]
#define OFF_RHI   12                       // [NB][3]
#define OFF_CNT   24                       // 8 ints: [0..3]=pred cnt, [4..7]=tgt cnt
#define OFF_PN    32                       // [NB][NPTS] |p|^2
#define OFF_TN    (OFF_PN  + NB*NPTS)      // [NB][NPTS] |t|^2
#define OFF_IP    (OFF_TN  + NB*NPTS)      // [NB][NPTS] pred indicator -> psel (in place)
#define OFF_IT    (OFF_IP  + NB*NPTS)      // [NB][NPTS] tgt indicator
#define OFF_DIFF  (OFF_IT  + NB*NPTS)      // [NB][NPTS] min sq NN dist
#define OFF_LOSSB (OFF_DIFF + NB*NPTS)     // [NB]
#define OFF_PK    (OFF_LOSSB + NB)         // [NB][TPAD] float4 {x,y,z,tn+(1-tsel)*BIG}
// OFF_PK = 140036 floats -> byte offset 560144, 16B aligned for b128 loads.

// ------------------------------------------------------------------
// K1: per-batch min/max bounds of pred -> r_lo/r_hi; zero counters.
// ------------------------------------------------------------------
__global__ void k_bounds(const float* __restrict__ pred, float* __restrict__ ws) {
    __shared__ float smn[3][256];
    __shared__ float smx[3][256];
    int b = blockIdx.x, t = threadIdx.x;
    const float* P = pred + (size_t)b * NPTS * 3;
    float mn0 = FLT_MAX, mn1 = FLT_MAX, mn2 = FLT_MAX;
    float mx0 = -FLT_MAX, mx1 = -FLT_MAX, mx2 = -FLT_MAX;
    for (int i = t; i < NPTS; i += 256) {
        float x = P[i*3+0], y = P[i*3+1], z = P[i*3+2];
        mn0 = fminf(mn0, x); mx0 = fmaxf(mx0, x);
        mn1 = fminf(mn1, y); mx1 = fmaxf(mx1, y);
        mn2 = fminf(mn2, z); mx2 = fmaxf(mx2, z);
    }
    smn[0][t]=mn0; smn[1][t]=mn1; smn[2][t]=mn2;
    smx[0][t]=mx0; smx[1][t]=mx1; smx[2][t]=mx2;
    __syncthreads();
    for (int s = 128; s > 0; s >>= 1) {
        if (t < s) {
            #pragma unroll
            for (int c = 0; c < 3; ++c) {
                smn[c][t] = fminf(smn[c][t], smn[c][t+s]);
                smx[c][t] = fmaxf(smx[c][t], smx[c][t+s]);
            }
        }
        __syncthreads();
    }
    if (t == 0) {
        const float bs[3] = {0.1f, 1.0f, 1.0f};
        const float bi[3] = {4.0f, 0.0f, 0.0f};   // BLOCK_I = 4, j = k = 0
        #pragma unroll
        for (int c = 0; c < 3; ++c) {
            float mn = smn[c][0], mx = smx[c][0];
            float w  = mx - mn;
            float lo = mn + 0.05f * w;
            float hi = mx - 0.05f * w;
            float rlo = (hi - lo) * bi[c] * bs[c] + lo;
            float rhi = rlo + (hi - lo) * bs[c];
            ws[OFF_RLO + b*3 + c] = rlo;
            ws[OFF_RHI + b*3 + c] = rhi;
        }
        int* cnt = (int*)(ws + OFF_CNT);
        cnt[b] = 0; cnt[NB + b] = 0;
    }
}

// ------------------------------------------------------------------
// K2a: indicators + squared norms + selected-point counts.
// grid.z: 0 = pred, 1 = tgt.
// ------------------------------------------------------------------
__global__ void k_select(const float* __restrict__ pred, const float* __restrict__ tgt,
                         float* __restrict__ ws) {
    __shared__ int scnt[256];
    int b = blockIdx.y, which = blockIdx.z;
    int t = threadIdx.x;
    int i = blockIdx.x * 256 + t;
    int indv = 0;
    if (i < NPTS) {
        const float* S = (which ? tgt : pred) + (size_t)b * NPTS * 3;
        float x = S[i*3+0], y = S[i*3+1], z = S[i*3+2];
        const float* rlo = ws + OFF_RLO + b*3;
        const float* rhi = ws + OFF_RHI + b*3;
        bool ind = (x > rlo[0]) && (x < rhi[0]) &&
                   (y > rlo[1]) && (y < rhi[1]) &&
                   (z > rlo[2]) && (z < rhi[2]);
        indv = ind ? 1 : 0;
        float nrm = x*x + y*y + z*z;
        if (which == 0) {
            ws[OFF_PN + b*NPTS + i] = nrm;
            ws[OFF_IP + b*NPTS + i] = (float)indv;
        } else {
            ws[OFF_TN + b*NPTS + i] = nrm;
            ws[OFF_IT + b*NPTS + i] = (float)indv;
        }
    }
    scnt[t] = indv;
    __syncthreads();
    for (int s = 128; s > 0; s >>= 1) {
        if (t < s) scnt[t] += scnt[t+s];
        __syncthreads();
    }
    if (t == 0) atomicAdd(((int*)(ws + OFF_CNT)) + which*NB + b, scnt[0]);
}

// ------------------------------------------------------------------
// K2b: apply >=500 fallback. For targets, pack the whole B-side into
// float4 {x,y,z, tn + (1-tsel)*BIG} per column, padded to TPAD cols.
// (Folding the mask into tn before the clamp cannot change the row min:
//  masked columns carry +BIG and never win; unmasked columns identical.)
// ------------------------------------------------------------------
__global__ void k_finalize_sel(const float* __restrict__ tgt, float* __restrict__ ws) {
    int b = blockIdx.y, which = blockIdx.z;
    int i = blockIdx.x * 256 + threadIdx.x;
    const int* cnt = (const int*)(ws + OFF_CNT);
    if (which == 0) {
        if (i >= NPTS) return;
        float v = ws[OFF_IP + b*NPTS + i];
        ws[OFF_IP + b*NPTS + i] = (cnt[b] >= 500) ? v : (i == 0 ? 1.0f : 0.0f);
    } else {
        if (i >= TPAD) return;
        float4* PKb = (float4*)(ws + OFF_PK) + (size_t)b * TPAD;
        if (i < NPTS) {
            float ind  = ws[OFF_IT + b*NPTS + i];
            float tsel = (cnt[NB + b] >= 500) ? ind : 1.0f;
            float tne  = ws[OFF_TN + b*NPTS + i] + (1.0f - tsel) * BIGC;
            const float* Tp = tgt + (size_t)b * NPTS * 3 + (size_t)i * 3;
            PKb[i] = make_float4(Tp[0], Tp[1], Tp[2], tne);
        } else {
            PKb[i] = make_float4(0.0f, 0.0f, 0.0f, 4.0e10f);  // pad col never wins
        }
    }
}

// ------------------------------------------------------------------
// K3: fused distance-matrix + row-min via V_WMMA_F32_16X16X4_F32.
// One wave32 per 16-row pred tile; sweep 440 packed target tiles.
// Inner loop tracks min_j (tne_j - 2*dot): since max(.,0) is monotone,
//   min_j max(pn + tne_j - 2 dot, 0) = max(pn + min_j(tne_j - 2 dot), 0)
// so pn and the clamp are applied once at the end, outside the loop.
// Inner loop: 1x global_load_b128 + 1x wmma + 8x {fma, min}.
// ------------------------------------------------------------------
__global__ void __launch_bounds__(32)
k_chamfer_min(const float* __restrict__ pred, float* __restrict__ ws) {
    int b    = blockIdx.y;
    int m0   = blockIdx.x * 16;
    int lane = threadIdx.x;
    int half = lane >> 4;        // 0: K=0,1 ; 1: K=2,3(pad)
    int l15  = lane & 15;

    const float*  P   = pred + (size_t)b * NPTS * 3;
    const float*  PN  = ws + OFF_PN + b*NPTS;
    const float4* PKb = (const float4*)(ws + OFF_PK) + (size_t)b * TPAD;
    float*       DIFF = ws + OFF_DIFF + b*NPTS;

    // A fragment (uniform across the whole j-sweep)
    int mrow = m0 + l15; if (mrow >= NPTS) mrow = NPTS - 1;
    float pa0 = P[mrow*3 + (half ? 2 : 0)];
    float pa1 = P[mrow*3 + 1];
    v2f a; a.x = pa0; a.y = half ? 0.0f : pa1;

    float rmin[8];
    #pragma unroll
    for (int r = 0; r < 8; ++r) rmin[r] = FLT_MAX;

    #pragma unroll 4
    for (int jt = 0; jt < TILES_I; ++jt) {
        float4 q = PKb[jt*16 + l15];           // {x, y, z, tn+(1-tsel)*BIG}
        v2f bb;
        bb.x = half ? q.z : q.x;
        bb.y = half ? 0.0f : q.y;

        v8f c = {};
        c = __builtin_amdgcn_wmma_f32_16x16x4_f32(
                /*neg_a=*/false, a, /*neg_b=*/false, bb,
                /*c_mod=*/(short)0, c, /*reuse_a=*/false, /*reuse_b=*/false);

        #pragma unroll
        for (int r = 0; r < 8; ++r) {
            float v = __builtin_fmaf(-2.0f, c[r], q.w);   // tne - 2*dot
            rmin[r] = fminf(rmin[r], v);
        }
    }

    // min across the 16 lanes of each half (columns), then apply pn + clamp
    #pragma unroll
    for (int r = 0; r < 8; ++r) {
        float v = rmin[r];
        #pragma unroll
        for (int off = 8; off >= 1; off >>= 1)
            v = fminf(v, __shfl_xor(v, off, 16));
        rmin[r] = v;
    }
    if (l15 == 0) {
        #pragma unroll
        for (int r = 0; r < 8; ++r) {
            int m = m0 + half*8 + r;
            if (m < NPTS) {
                float pn = PN[m];
                DIFF[m] = fmaxf(pn + rmin[r], 0.0f);   // = min_j max(dist,0)+mask
            }
        }
    }
}

// ------------------------------------------------------------------
// K4: per batch — exact kth value via LDS bitonic sort, then kept sums.
// ------------------------------------------------------------------
__global__ void k_loss(const float* __restrict__ mask, float* __restrict__ ws) {
    __shared__ float s[NSORT];
    int b = blockIdx.x, t = threadIdx.x;
    const float* diff = ws + OFF_DIFF + b*NPTS;
    const float* psel = ws + OFF_IP   + b*NPTS;
    const int*   cnt  = (const int*)(ws + OFF_CNT);

    for (int i = t; i < NSORT; i += 256) {
        float v = FLT_MAX;
        if (i < NPTS) v = (psel[i] > 0.0f) ? diff[i] : BIGC;
        s[i] = v;
    }
    __syncthreads();

    // bitonic sort ascending
    for (int kk = 2; kk <= NSORT; kk <<= 1) {
        for (int j = kk >> 1; j > 0; j >>= 1) {
            for (int i = t; i < NSORT; i += 256) {
                int ixj = i ^ j;
                if (ixj > i) {
                    float x = s[i], y = s[ixj];
                    bool up = ((i & kk) == 0);
                    if ((x > y) == up) { s[i] = y; s[ixj] = x; }
                }
            }
            __syncthreads();
        }
    }

    int n_sel = (cnt[b] >= 500) ? cnt[b] : 1;     // sum(psel)
    int k = 1 + (n_sel >> 1);                     // 1 + floor(n_sel*0.5)
    float thr = s[k - 1];
    __syncthreads();

    float sumsq = 0.0f, summ = 0.0f;
    const float* m = mask + b*NPTS;
    for (int i = t; i < NPTS; i += 256) {
        float ds = (psel[i] > 0.0f) ? diff[i] : BIGC;
        if (ds < thr) {                           // strict <
            float d = diff[i], mm = m[i];
            sumsq += d * d * mm;                  // diff squared again, per torch code
            summ  += mm;
        }
    }
    s[t] = sumsq; s[256 + t] = summ;
    __syncthreads();
    for (int sd = 128; sd > 0; sd >>= 1) {
        if (t < sd) { s[t] += s[t+sd]; s[256+t] += s[256+t+sd]; }
        __syncthreads();
    }
    if (t == 0) ws[OFF_LOSSB + b] = s[0] / (s[256] + 1e-12f);
}

// ------------------------------------------------------------------
// K5: batch mean + focal/alpha wrapper (gamma=0 => fw = 1/(1+1e-12)).
// ------------------------------------------------------------------
__global__ void k_final(const float* __restrict__ alpha, const float* __restrict__ ws,
                        float* __restrict__ out) {
    float loss = 0.0f;
    #pragma unroll
    for (int b = 0; b < NB; ++b) loss += ws[OFF_LOSSB + b];
    loss *= (1.0f / NB);
    float a  = alpha[0];
    float x  = __expf(-a) * loss;   // exp(-alpha) * loss
    float fw = 1.0f;                // x ** FOCAL_GAMMA (gamma = 0)
    fw = fw / (fw + 1e-12f);
    out[0] = 1.0f * (fw * x + a);   // LOSS_WEIGHT = 1
}

// ------------------------------------------------------------------
extern "C" void kernel_launch(void* const* d_in, const int* in_sizes, int n_in,
                              void* d_out, int out_size, void* d_ws, size_t ws_size,
                              hipStream_t stream) {
    (void)in_sizes; (void)n_in; (void)out_size; (void)ws_size;
    const float* pred  = (const float*)d_in[0];
    const float* tgt   = (const float*)d_in[1];
    const float* mask  = (const float*)d_in[2];
    const float* alpha = (const float*)d_in[3];
    float* ws  = (float*)d_ws;
    float* out = (float*)d_out;

    dim3 gsel((TPAD + 255) / 256, NB, 2);   // covers both NPTS and TPAD ranges

    k_bounds      <<<dim3(NB),          dim3(256), 0, stream>>>(pred, ws);
    k_select      <<<gsel,              dim3(256), 0, stream>>>(pred, tgt, ws);
    k_finalize_sel<<<gsel,              dim3(256), 0, stream>>>(tgt, ws);
    k_chamfer_min <<<dim3(TILES_M, NB), dim3(32),  0, stream>>>(pred, ws);
    k_loss        <<<dim3(NB),          dim3(256), 0, stream>>>(mask, ws);
    k_final       <<<dim3(1),           dim3(1),   0, stream>>>(alpha, ws, out);
}